// ApproximatePVLFM_42880953483820
// MI455X (gfx1250) — compile-verified
//
#include <hip/hip_runtime.h>
#include <hip/hip_bf16.h>

// ---------------- problem constants ----------------
constexpr int S      = 1024;
constexpr int L      = 8192;
constexpr int F      = 3 * (L - 1) + 1;   // 24574
constexpr int NSTEP  = L - 1;             // 8191 RK steps
constexpr int NSEG   = 64;                // segments for parallel linear scan
constexpr int SEGLEN = 128;               // NSEG*SEGLEN >= NSTEP

// ws layout (float offsets)
constexpr size_t OFF_HS   = 0;                         // L*S floats
constexpr size_t OFF_COEF = (size_t)L * S;             // float4[L]
constexpr size_t OFF_P    = OFF_COEF + 4ull * L;       // [NSEG]
constexpr size_t OFF_Q    = OFF_P + 64;                // [NSEG*S]
constexpr size_t OFF_HB   = OFF_Q + (size_t)NSEG * S;  // [NSEG*S]
constexpr size_t OFF_VAR  = OFF_HB + (size_t)NSEG * S; // [L]
constexpr size_t OFF_D0   = OFF_VAR + L;               // [1]

typedef __attribute__((ext_vector_type(2))) float v2f;
typedef __attribute__((ext_vector_type(4))) float f4v;
typedef __attribute__((ext_vector_type(8))) float v8f;

__device__ __forceinline__ float interval_dev(float raw, float lo, float hi) {
    return lo + (hi - lo) / (1.0f + expf(-raw));
}

// ---------------- kernel: per-step affine coefficients ----------------
// h' = A*h + C1*f[3t+1] + C2*f[3t+2] + C3*f[3t+3]   (+ D0*f[0] at t==0)
__global__ void k_coeff(const float* __restrict t_f, const float* __restrict ra,
                        const float* __restrict rb, const float* __restrict rc,
                        float4* __restrict coef, float* __restrict d0_out) {
    int t = blockIdx.x * blockDim.x + threadIdx.x;
    if (t >= NSTEP) return;
    float a = interval_dev(ra[0], 0.0001f, 1.0f);
    float b = interval_dev(rb[0], 0.001f, 1.0f);
    float c = interval_dev(rc[0], 0.001f, 1.0f);
    float t0 = t_f[t];
    float dt = t_f[t + 1] - t0;
    const float pi = 3.14159265358979323846f;
    float cp = c * pi;
    float s1 = b * sinf(cp * t0);
    float s2 = b * sinf(cp * (t0 + dt * (1.0f / 3.0f)));
    float s3 = b * sinf(cp * (t0 + dt * (2.0f / 3.0f)));
    float s4 = b * sinf(cp * (t0 + dt));
    // homogeneous part: k_i = u_i + v_i*h
    float v1 = -s1;
    float v2 = -s2 * (1.0f + dt * v1 * (1.0f / 3.0f));
    float v3 = -s3 * (1.0f + dt * (v2 - v1 * (1.0f / 3.0f)));
    float v4 = -s4 * (1.0f + dt * (v1 - v2 + v3));
    float A  = 1.0f + dt * (v1 + 3.0f * (v2 + v3) + v4) * 0.125f;
    // forcing part: coefficients of g_j = a*f_j inside u_i
    float p21 = -s2 * dt * (1.0f / 3.0f);
    float p31 = -s3 * dt * (p21 - (1.0f / 3.0f));
    float p32 = -s3 * dt;
    float p41 = -s4 * dt * (1.0f - p21 + p31);
    float p42 = -s4 * dt * (p32 - 1.0f);
    float p43 = -s4 * dt;
    float k  = a * dt * 0.125f;
    float B1 = k * (1.0f + 3.0f * p21 + 3.0f * p31 + p41);
    float B2 = k * (3.0f + 3.0f * p32 + p42);
    float B3 = k * (3.0f + p43);
    float B4 = k;
    float4 cf;
    cf.x = A;
    if (t == 0) { cf.y = B2; cf.z = B3; cf.w = B4; d0_out[0] = B1; } // f idx {1,2,3}, D0*f[0]
    else        { cf.y = B1 + B2; cf.z = B3; cf.w = B4; }           // k1,k2 share f[3t+1]
    coef[t] = cf;
}

// ---------------- kernel: per-segment product of A ----------------
__global__ void k_pseg(const float4* __restrict coef, float* __restrict P) {
    int seg = threadIdx.x;
    if (seg >= NSEG) return;
    int t0 = seg * SEGLEN;
    int t1 = t0 + SEGLEN; if (t1 > NSTEP) t1 = NSTEP;
    float p = 1.0f;
    for (int t = t0; t < t1; ++t) p *= coef[t].x;
    P[seg] = p;
}

// ---------------- kernel: phase A — forced response per (series, segment) ----------------
__global__ void k_phaseA(const float* __restrict f, const float4* __restrict coef,
                         const float* __restrict d0p, float* __restrict q) {
    int gid = blockIdx.x * blockDim.x + threadIdx.x;   // NSEG*S items
    int s   = gid & (S - 1);
    int seg = gid >> 10;                               // uniform per block (256 | 1024)
    const float* frow = f + (size_t)s * F;
    int t0 = seg * SEGLEN;
    int t1 = t0 + SEGLEN; if (t1 > NSTEP) t1 = NSTEP;
    float d0 = d0p[0];
    float h = 0.0f;                                    // forced response from zero state
    for (int t = t0; t < t1; ++t) {
        float4 cf = coef[t];
        const float* fp = frow + 3 * t + 1;
        h = fmaf(cf.x, h, fmaf(cf.y, fp[0], fmaf(cf.z, fp[1], cf.w * fp[2])));
        if (t == 0) h = fmaf(d0, frow[0], h);
    }
    q[(size_t)seg * S + s] = h;
}

// ---------------- kernel: phase B — boundary scan over segments ----------------
__global__ void k_phaseB(const float* __restrict h0, const float* __restrict P,
                         const float* __restrict q, float* __restrict hb) {
    int s = blockIdx.x * blockDim.x + threadIdx.x;
    if (s >= S) return;
    float h = h0[s];
    hb[s] = h;
    for (int seg = 0; seg < NSEG - 1; ++seg) {
        h = fmaf(P[seg], h, q[(size_t)seg * S + s]);
        hb[(size_t)(seg + 1) * S + s] = h;
    }
}

// ---------------- kernel: phase C — replay segments, emit hs[t*S+s] ----------------
__global__ void k_phaseC(const float* __restrict f, const float4* __restrict coef,
                         const float* __restrict d0p, const float* __restrict hb,
                         float* __restrict hs) {
    int gid = blockIdx.x * blockDim.x + threadIdx.x;
    int s   = gid & (S - 1);
    int seg = gid >> 10;
    const float* frow = f + (size_t)s * F;
    int t0 = seg * SEGLEN;
    int t1 = t0 + SEGLEN; if (t1 > NSTEP) t1 = NSTEP;
    float d0 = d0p[0];
    float h = hb[(size_t)seg * S + s];
    if (seg == 0) hs[s] = h;                           // t = 0 row (= h0)
    for (int t = t0; t < t1; ++t) {
        float4 cf = coef[t];
        const float* fp = frow + 3 * t + 1;
        h = fmaf(cf.x, h, fmaf(cf.y, fp[0], fmaf(cf.z, fp[1], cf.w * fp[2])));
        if (t == 0) h = fmaf(d0, frow[0], h);
        hs[(size_t)(t + 1) * S + s] = h;               // coalesced across lanes
    }
}

// ---------------- kernel: WMMA reduction over S (mean + ddof-1 variance) ----------------
// D = ones(16x4) x B(4x16) + C  => every D element = column sum of B.
// Column n of each wave's tile = timestep tbase+n; K rows = 4 consecutive series.
// B layout (ISA 7.12.2): lanes 0-15 hold K={0,1} in vgpr{0,1}; lanes 16-31 hold K={2,3}.
__global__ void k_reduce(const float* __restrict hs, const float* __restrict eps,
                         const float* __restrict raw_noise,
                         float* __restrict hmean_out, float* __restrict hvar_out) {
    int wave = threadIdx.x >> 5;                 // 8 waves / block
    int lane = threadIdx.x & 31;
    int tbase = (blockIdx.x * 8 + wave) * 16;    // 16 timesteps per wave
    int n  = lane & 15;
    int hi = lane >> 4;
    float noise = interval_dev(raw_noise[0], 0.05f, 0.5f);
    const float* hrow = hs  + (size_t)(tbase + n) * S + 2 * hi;
    const float* erow = eps + (size_t)(tbase + n) * S + 2 * hi;
    v2f ones; ones[0] = 1.0f; ones[1] = 1.0f;    // A = all-ones 16x4 (layout-invariant)
    v8f ch  = {0.f,0.f,0.f,0.f,0.f,0.f,0.f,0.f};
    v8f cx  = {0.f,0.f,0.f,0.f,0.f,0.f,0.f,0.f};
    v8f cx2 = {0.f,0.f,0.f,0.f,0.f,0.f,0.f,0.f};
    for (int s0 = 0; s0 < S; s0 += 4) {
        float hv0 = hrow[s0], hv1 = hrow[s0 + 1];
        float e0  = erow[s0], e1  = erow[s0 + 1];
        float x0 = fmaf(noise, e0, hv0);
        float x1 = fmaf(noise, e1, hv1);
        v2f bh;  bh[0]  = hv0;     bh[1]  = hv1;
        v2f bx;  bx[0]  = x0;      bx[1]  = x1;
        v2f bx2; bx2[0] = x0 * x0; bx2[1] = x1 * x1;
        ch  = __builtin_amdgcn_wmma_f32_16x16x4_f32(false, ones, false, bh,  (short)0, ch,  false, false);
        cx  = __builtin_amdgcn_wmma_f32_16x16x4_f32(false, ones, false, bx,  (short)0, cx,  false, false);
        cx2 = __builtin_amdgcn_wmma_f32_16x16x4_f32(false, ones, false, bx2, (short)0, cx2, false, false);
    }
    if (lane < 16) {
        float sh = ch[0], sx = cx[0], sx2 = cx2[0];
        hmean_out[tbase + n] = sh * (1.0f / S);
        hvar_out[tbase + n]  = (sx2 - sx * sx * (1.0f / S)) * (1.0f / (S - 1));
    }
}

// ---------------- kernel: nontemporal zero-fill of covar (268 MB) ----------------
__global__ void k_fill(f4v* __restrict covar4, size_t n4) {
    size_t i = (size_t)blockIdx.x * blockDim.x + threadIdx.x;
    size_t stride = (size_t)gridDim.x * blockDim.x;
    f4v z = {0.f, 0.f, 0.f, 0.f};
    for (; i < n4; i += stride) __builtin_nontemporal_store(z, covar4 + i);
}

// ---------------- kernel: diagonal = var + 1e-4 ----------------
__global__ void k_diag(float* __restrict covar, const float* __restrict hvar) {
    int i = blockIdx.x * blockDim.x + threadIdx.x;
    if (i < L) covar[(size_t)i * (L + 1)] = hvar[i] + 0.0001f;
}

extern "C" void kernel_launch(void* const* d_in, const int* in_sizes, int n_in,
                              void* d_out, int out_size, void* d_ws, size_t ws_size,
                              hipStream_t stream) {
    const float* t_f       = (const float*)d_in[0];
    const float* f         = (const float*)d_in[1];
    const float* h0        = (const float*)d_in[2];
    const float* raw_a     = (const float*)d_in[3];
    const float* raw_b     = (const float*)d_in[4];
    const float* raw_c     = (const float*)d_in[5];
    const float* raw_noise = (const float*)d_in[6];
    const float* eps       = (const float*)d_in[7];

    float*  out   = (float*)d_out;
    float*  covar = out + L;                    // h_mean first, then LxL covar
    float*  w     = (float*)d_ws;
    float*  hs    = w + OFF_HS;
    float4* coef  = (float4*)(w + OFF_COEF);
    float*  P     = w + OFF_P;
    float*  q     = w + OFF_Q;
    float*  hb    = w + OFF_HB;
    float*  hvar  = w + OFF_VAR;
    float*  d0    = w + OFF_D0;

    // Big store first (independent of everything except the diagonal pass).
    size_t n4 = ((size_t)L * (size_t)L) / 4;
    k_fill<<<8192, 256, 0, stream>>>((f4v*)covar, n4);

    // Per-step affine coefficients, then segmented parallel linear scan.
    k_coeff<<<(NSTEP + 255) / 256, 256, 0, stream>>>(t_f, raw_a, raw_b, raw_c, coef, d0);
    k_pseg<<<1, 64, 0, stream>>>(coef, P);
    k_phaseA<<<(NSEG * S) / 256, 256, 0, stream>>>(f, coef, d0, q);
    k_phaseB<<<S / 256, 256, 0, stream>>>(h0, P, q, hb);
    k_phaseC<<<(NSEG * S) / 256, 256, 0, stream>>>(f, coef, d0, hb, hs);

    // WMMA-based mean/variance over S; mean goes straight to d_out[0..L).
    k_reduce<<<L / 128, 256, 0, stream>>>(hs, eps, raw_noise, out, hvar);

    // Diagonal after the zero-fill.
    k_diag<<<L / 256, 256, 0, stream>>>(covar, hvar);
}